// CrossMamba2Simple_15745350107168
// MI455X (gfx1250) — compile-verified
//
#include <hip/hip_runtime.h>
#include <hip/hip_bf16.h>

// ---------------- model constants ----------------
#define D_MODEL   192
#define D_STATE   16
#define D_CONV    4
#define HEADDIM   16
#define D_INNER   384
#define NHEADS    24
#define CHUNK     256
#define BATCH     2
#define SEQLEN    8192
#define D_IN_PROJ 808          // 2*384 + 16 + 24
#define CONV_DIM  400          // 384 + 16
#define NCHUNKS_TOTAL 64       // BATCH * SEQLEN/CHUNK
#define ROWS      16384        // BATCH * SEQLEN

typedef float v2f __attribute__((ext_vector_type(2)));
typedef float v8f __attribute__((ext_vector_type(8)));

static __device__ __forceinline__ v8f wmma4(v2f a, v2f b, v8f c) {
  // V_WMMA_F32_16X16X4_F32 : D(16x16 f32) = A(16x4 f32) * B(4x16 f32) + C
  return __builtin_amdgcn_wmma_f32_16x16x4_f32(false, a, false, b, (short)0, c,
                                               false, false);
}

static __device__ __forceinline__ float sigmoidf_(float x) {
  return 1.0f / (1.0f + __expf(-x));
}

// ---------------------------------------------------------------------------
// Generic batched GEMM: out[M,N] = A[M,K] @ W[N,K]^T   (fp32 WMMA 16x16x4)
// One 16x16 tile per wave; 8 waves per block. M,K multiples of 4/16; N padded.
// ---------------------------------------------------------------------------
__global__ __launch_bounds__(256)
void gemm_wmma_nt(const float* __restrict__ A, const float* __restrict__ W,
                  float* __restrict__ out, int M, int N, int K,
                  int lda, int ldw, int ldo,
                  long long sA, long long sW, long long sO)
{
  long long bz = blockIdx.z;
  A   += bz * sA;
  W   += bz * sW;
  out += bz * sO;

  int Mt = M >> 4;
  int Nt = (N + 15) >> 4;
  int tile = blockIdx.x * 8 + (threadIdx.x >> 5);
  if (tile >= Mt * Nt) return;

  int lane = threadIdx.x & 31;
  int hi = lane >> 4;          // lane half selects K pair
  int lo = lane & 15;
  int tm = tile / Nt;
  int tn = tile - tm * Nt;

  int m = tm * 16 + lo;                 // A row for this lane
  int n = tn * 16 + lo;                 // W row (B column) for this lane
  int nc = (n < N) ? n : (N - 1);       // clamp for ragged N (store is masked)

  const float* Ap = A + (long long)m  * lda;
  const float* Wp = W + (long long)nc * ldw;

  v8f acc = {0.f, 0.f, 0.f, 0.f, 0.f, 0.f, 0.f, 0.f};
  for (int k = 0; k < K; k += 4) {
    v2f a, b;
    a.x = Ap[k + hi * 2];
    a.y = Ap[k + hi * 2 + 1];
    b.x = Wp[k + hi * 2];
    b.y = Wp[k + hi * 2 + 1];
    acc = wmma4(a, b, acc);
  }

  if (n < N) {
#pragma unroll
    for (int v = 0; v < 8; v++) {
      int mr = v + hi * 8;  // D row = vgpr + (lane>=16)*8
      out[(long long)(tm * 16 + mr) * ldo + n] = acc[v];
    }
  }
}

// ---------------------------------------------------------------------------
// Fused causal depthwise conv (width 4) + SiLU, plus softplus(dt).
// One thread per (b, l, channel) with 440 channels: 384 x | 16 B | 16 C | 24 dt
// ---------------------------------------------------------------------------
__global__ __launch_bounds__(256)
void conv_dt_kernel(const float* __restrict__ zx,   // (ROWS, 808)
                    const float* __restrict__ cp,   // (ROWS, 16)
                    const float* __restrict__ conv_w,   // (400,1,4)
                    const float* __restrict__ conv_b,   // (400)
                    const float* __restrict__ conv_w_b, // (16,1,4)
                    const float* __restrict__ conv_b_b, // (16)
                    const float* __restrict__ dt_bias,  // (24)
                    float* __restrict__ x,    // (ROWS, 384)
                    float* __restrict__ Bv,   // (ROWS, 16)
                    float* __restrict__ Cv,   // (ROWS, 16)
                    float* __restrict__ dtb)  // (ROWS, 24)
{
  long long gid = (long long)blockIdx.x * blockDim.x + threadIdx.x;
  const long long total = (long long)ROWS * 440;
  if (gid >= total) return;
  int ch = (int)(gid % 440);
  long long bl = gid / 440;            // flat (b*SEQLEN + l)
  int l = (int)(bl & (SEQLEN - 1));    // position within sequence

  if (ch < CONV_DIM) {                 // conv over zxbdt cols [384, 784)
    float acc = conv_b[ch];
#pragma unroll
    for (int j = 0; j < D_CONV; j++) {
      int tt = l - (D_CONV - 1) + j;
      if (tt >= 0) {
        float v = zx[(bl + (long long)(tt - l)) * D_IN_PROJ + D_INNER + ch];
        acc += conv_w[ch * D_CONV + j] * v;
      }
    }
    float s = acc * sigmoidf_(acc);
    if (ch < D_INNER) x[bl * D_INNER + ch] = s;
    else              Bv[bl * D_STATE + (ch - D_INNER)] = s;
  } else if (ch < CONV_DIM + D_STATE) { // conv over support projection
    int cc = ch - CONV_DIM;
    float acc = conv_b_b[cc];
#pragma unroll
    for (int j = 0; j < D_CONV; j++) {
      int tt = l - (D_CONV - 1) + j;
      if (tt >= 0) {
        float v = cp[(bl + (long long)(tt - l)) * D_STATE + cc];
        acc += conv_w_b[cc * D_CONV + j] * v;
      }
    }
    Cv[bl * D_STATE + cc] = acc * sigmoidf_(acc);
  } else {                              // softplus(dt)
    int hh = ch - (CONV_DIM + D_STATE);
    float v = zx[bl * D_IN_PROJ + (D_IN_PROJ - NHEADS) + hh] + dt_bias[hh];
    float sp = (v > 20.f) ? v : log1pf(__expf(v));
    dtb[bl * NHEADS + hh] = sp;
  }
}

// ---------------------------------------------------------------------------
// Per (chunk, head): cumsum(a), Y_diag = (G . L) @ xdt  (WMMA, K=256),
// chunk_state = sum_s B[s]^T decay[s] xdt[s]  (WMMA, cross-wave LDS reduce).
// grid = (64, 24), block = 256 (8 waves).
// ---------------------------------------------------------------------------
__global__ __launch_bounds__(256)
void chunk_kernel(const float* __restrict__ x,     // (ROWS,384)
                  const float* __restrict__ Bv,    // (ROWS,16)
                  const float* __restrict__ dtb,   // (ROWS,24)
                  const float* __restrict__ Gws,   // (64,256,256)
                  const float* __restrict__ A_log, // (24)
                  float* __restrict__ Ydiag,       // (ROWS,384)
                  float* __restrict__ acum_ws,     // (64,24,256)
                  float* __restrict__ states_ws)   // (64,24,16,16)
{
  int bc = blockIdx.x;
  int h  = blockIdx.y;
  int tid = threadIdx.x;
  int lane = tid & 31;
  int w = tid >> 5;
  int hi = lane >> 4;
  int lo = lane & 15;

  __shared__ float s_ac[CHUNK];            // cumsum(dt*A)
  __shared__ float s_xdt[CHUNK * HEADDIM]; // x*dt for this head
  __shared__ float s_B[CHUNK * D_STATE];   // B values
  __shared__ float s_red[8 * 32 * 8];      // cross-wave state reduction

  long long row = (long long)bc * CHUNK + tid;
  float dtv = dtb[row * NHEADS + h];
  float Ah = -__expf(A_log[h]);
#pragma unroll
  for (int p = 0; p < 16; p++) {
    s_xdt[tid * 16 + p] = x[row * D_INNER + h * 16 + p] * dtv;
    s_B[tid * 16 + p]   = Bv[row * D_STATE + p];
  }
  s_ac[tid] = dtv * Ah;
  __syncthreads();

  // Hillis-Steele inclusive scan over 256 positions
  for (int off = 1; off < CHUNK; off <<= 1) {
    float v = (tid >= off) ? s_ac[tid - off] : 0.f;
    __syncthreads();
    s_ac[tid] += v;
    __syncthreads();
  }
  float a_last = s_ac[CHUNK - 1];
  acum_ws[((long long)bc * NHEADS + h) * CHUNK + tid] = s_ac[tid];

  const float* G = Gws + (long long)bc * (CHUNK * CHUNK);

  // ---- Y_diag: 16 l-tiles, each wave owns 2; K loop over s-tiles <= l-tile
  for (int lt = w; lt < 16; lt += 8) {
    int l = lt * 16 + lo;
    float acl = s_ac[l];
    v8f acc = {0.f, 0.f, 0.f, 0.f, 0.f, 0.f, 0.f, 0.f};
    for (int st = 0; st <= lt; st++) {
#pragma unroll
      for (int kk = 0; kk < 4; kk++) {
        int s0 = st * 16 + kk * 4 + hi * 2;
        v2f a, b;
        float g0 = G[l * CHUNK + s0];
        float g1 = G[l * CHUNK + s0 + 1];
        a.x = (s0     <= l) ? g0 * __expf(acl - s_ac[s0])     : 0.f;
        a.y = (s0 + 1 <= l) ? g1 * __expf(acl - s_ac[s0 + 1]) : 0.f;
        b.x = s_xdt[s0 * 16 + lo];
        b.y = s_xdt[(s0 + 1) * 16 + lo];
        acc = wmma4(a, b, acc);
      }
    }
#pragma unroll
    for (int v = 0; v < 8; v++) {
      int mr = v + hi * 8;
      long long gr = (long long)bc * CHUNK + lt * 16 + mr;
      Ydiag[gr * D_INNER + h * 16 + lo] = acc[v];
    }
  }

  // ---- chunk state: each wave covers s in [w*32, w*32+32)
  v8f sacc = {0.f, 0.f, 0.f, 0.f, 0.f, 0.f, 0.f, 0.f};
#pragma unroll
  for (int kk = 0; kk < 8; kk++) {
    int s0 = w * 32 + kk * 4 + hi * 2;
    v2f a, b;
    a.x = s_B[s0 * 16 + lo]       * __expf(a_last - s_ac[s0]);
    a.y = s_B[(s0 + 1) * 16 + lo] * __expf(a_last - s_ac[s0 + 1]);
    b.x = s_xdt[s0 * 16 + lo];
    b.y = s_xdt[(s0 + 1) * 16 + lo];
    sacc = wmma4(a, b, sacc);
  }
  __syncthreads();
#pragma unroll
  for (int v = 0; v < 8; v++) s_red[w * 256 + lane * 8 + v] = sacc[v];
  __syncthreads();
  {
    int v  = tid & 7;
    int ln = tid >> 3;  // source lane 0..31
    float sum = 0.f;
#pragma unroll
    for (int ww = 0; ww < 8; ww++) sum += s_red[ww * 256 + ln * 8 + v];
    int mr = v + (ln >> 4) * 8;  // state index n
    int nc = ln & 15;            // headdim index p
    states_ws[((long long)bc * NHEADS + h) * 256 + mr * 16 + nc] = sum;
  }
}

// ---------------------------------------------------------------------------
// Inter-chunk state recurrence. grid = BATCH*NHEADS = 48, block = 256.
// prev[z] = running state BEFORE chunk z;  S = S*exp(a_chunk) + chunk_state
// ---------------------------------------------------------------------------
__global__ __launch_bounds__(256)
void scan_kernel(const float* __restrict__ states,  // (64,24,256)
                 const float* __restrict__ acum,    // (64,24,256)
                 float* __restrict__ prev)          // (64,24,256)
{
  int bh = blockIdx.x;
  int b = bh / NHEADS;
  int h = bh - b * NHEADS;
  int e = threadIdx.x;
  float S = 0.f;
  for (int z = 0; z < SEQLEN / CHUNK; z++) {
    long long bc = (long long)b * (SEQLEN / CHUNK) + z;
    long long base = (bc * NHEADS + h) * 256;
    prev[base + e] = S;
    float al = acum[base + 255];
    S = S * __expf(al) + states[base + e];
  }
}

// ---------------------------------------------------------------------------
// Y_off + D*x + SiLU gate + RMSNorm. grid = 64, block = 256 (thread = row).
// yg may alias Ydiag (read-before-write, same index, same thread).
// yn may alias x (x fully read in pass 1 for this thread's row).
// ---------------------------------------------------------------------------
__global__ __launch_bounds__(256)
void yoff_norm_kernel(const float* __restrict__ zx,    // (ROWS,808) gate z
                      const float* x,                   // (ROWS,384)
                      const float* __restrict__ Cv,     // (ROWS,16)
                      const float* Ydiag,               // (ROWS,384)
                      const float* __restrict__ prev,   // (64,24,256)
                      const float* __restrict__ acum,   // (64,24,256)
                      const float* __restrict__ Dv,     // (24)
                      const float* __restrict__ norm_w, // (384)
                      float* yg,                        // (ROWS,384) = Ydiag
                      float* yn)                        // (ROWS,384) = x
{
  __shared__ float sS[NHEADS * 256];  // 24KB: prefix states for this chunk
  int bc = blockIdx.x;
  int t = threadIdx.x;
  for (int i = t; i < NHEADS * 256; i += 256)
    sS[i] = prev[(long long)bc * (NHEADS * 256) + i];
  __syncthreads();

  long long row = (long long)bc * CHUNK + t;
  float cv[16];
#pragma unroll
  for (int n = 0; n < 16; n++) cv[n] = Cv[row * D_STATE + n];

  float sumsq = 0.f;
  for (int h = 0; h < NHEADS; h++) {
    float eA = __expf(acum[((long long)bc * NHEADS + h) * CHUNK + t]);
    float Dh = Dv[h];
#pragma unroll
    for (int p = 0; p < 16; p++) {
      float off = 0.f;
#pragma unroll
      for (int n = 0; n < 16; n++) off += cv[n] * sS[h * 256 + n * 16 + p];
      int c = h * 16 + p;
      float y = Ydiag[row * D_INNER + c] + eA * off + Dh * x[row * D_INNER + c];
      float zz = zx[row * D_IN_PROJ + c];
      float g = y * zz * sigmoidf_(zz);
      sumsq += g * g;
      yg[row * D_INNER + c] = g;
    }
  }
  float scale = rsqrtf(sumsq * (1.0f / (float)D_INNER) + 1e-5f);
  for (int c = 0; c < D_INNER; c++)
    yn[row * D_INNER + c] = yg[row * D_INNER + c] * scale * norm_w[c];
}

// ---------------------------------------------------------------------------
extern "C" void kernel_launch(void* const* d_in, const int* in_sizes, int n_in,
                              void* d_out, int out_size, void* d_ws, size_t ws_size,
                              hipStream_t stream) {
  const float* u        = (const float*)d_in[0];
  const float* support  = (const float*)d_in[1];
  const float* W_in     = (const float*)d_in[2];
  const float* W_in_b   = (const float*)d_in[3];
  const float* conv_w   = (const float*)d_in[4];
  const float* conv_b   = (const float*)d_in[5];
  const float* conv_w_b = (const float*)d_in[6];
  const float* conv_b_b = (const float*)d_in[7];
  const float* dt_bias  = (const float*)d_in[8];
  const float* A_log    = (const float*)d_in[9];
  const float* Dv       = (const float*)d_in[10];
  const float* norm_w   = (const float*)d_in[11];
  const float* W_out    = (const float*)d_in[12];
  float* out = (float*)d_out;

  // workspace layout (floats)
  float* ws = (float*)d_ws;
  float* zx   = ws;                              // ROWS*808
  float* cp   = zx  + (long long)ROWS * D_IN_PROJ;   // ROWS*16
  float* x    = cp  + (long long)ROWS * D_STATE;     // ROWS*384
  float* Bv   = x   + (long long)ROWS * D_INNER;     // ROWS*16
  float* Cv   = Bv  + (long long)ROWS * D_STATE;     // ROWS*16
  float* dtb  = Cv  + (long long)ROWS * D_STATE;     // ROWS*24
  float* G    = dtb + (long long)ROWS * NHEADS;      // 64*256*256
  float* acum = G    + (long long)NCHUNKS_TOTAL * CHUNK * CHUNK; // 64*24*256
  float* st   = acum + (long long)NCHUNKS_TOTAL * NHEADS * CHUNK;
  float* pv   = st   + (long long)NCHUNKS_TOTAL * NHEADS * 256;
  float* yd   = pv   + (long long)NCHUNKS_TOTAL * NHEADS * 256;  // ROWS*384
  float* yg   = yd;   // alias: read-before-write per element
  float* yn   = x;    // alias: x fully consumed before yn written

  // 1) zxbdt = u @ W_in^T         (16384 x 808, K=192)
  {
    int Mt = ROWS / 16, Nt = (D_IN_PROJ + 15) / 16;
    int blocks = (Mt * Nt + 7) / 8;
    gemm_wmma_nt<<<dim3(blocks, 1, 1), 256, 0, stream>>>(
        u, W_in, zx, ROWS, D_IN_PROJ, D_MODEL, D_MODEL, D_MODEL, D_IN_PROJ,
        0LL, 0LL, 0LL);
  }
  // 2) cproj = support @ W_in_b^T (16384 x 16, K=192)
  {
    int Mt = ROWS / 16, Nt = 1;
    int blocks = (Mt * Nt + 7) / 8;
    gemm_wmma_nt<<<dim3(blocks, 1, 1), 256, 0, stream>>>(
        support, W_in_b, cp, ROWS, D_STATE, D_MODEL, D_MODEL, D_MODEL, D_STATE,
        0LL, 0LL, 0LL);
  }
  // 3) conv + silu + softplus(dt)
  {
    long long total = (long long)ROWS * 440;
    int blocks = (int)((total + 255) / 256);
    conv_dt_kernel<<<dim3(blocks, 1, 1), 256, 0, stream>>>(
        zx, cp, conv_w, conv_b, conv_w_b, conv_b_b, dt_bias, x, Bv, Cv, dtb);
  }
  // 4) G[bc] = Cv_chunk @ Bv_chunk^T  (64 batches of 256x256, K=16)
  {
    int Mt = CHUNK / 16, Nt = CHUNK / 16;
    int blocks = (Mt * Nt + 7) / 8;
    gemm_wmma_nt<<<dim3(blocks, 1, NCHUNKS_TOTAL), 256, 0, stream>>>(
        Cv, Bv, G, CHUNK, CHUNK, D_STATE, D_STATE, D_STATE, CHUNK,
        (long long)CHUNK * D_STATE, (long long)CHUNK * D_STATE,
        (long long)CHUNK * CHUNK);
  }
  // 5) per-(chunk, head): cumsum, Y_diag, chunk states
  chunk_kernel<<<dim3(NCHUNKS_TOTAL, NHEADS, 1), 256, 0, stream>>>(
      x, Bv, dtb, G, A_log, yd, acum, st);
  // 6) inter-chunk scan
  scan_kernel<<<dim3(BATCH * NHEADS, 1, 1), 256, 0, stream>>>(st, acum, pv);
  // 7) Y_off + gate + RMSNorm
  yoff_norm_kernel<<<dim3(NCHUNKS_TOTAL, 1, 1), 256, 0, stream>>>(
      zx, x, Cv, yd, pv, acum, Dv, norm_w, yg, yn);
  // 8) out = yn @ W_out^T  (16384 x 192, K=384)
  {
    int Mt = ROWS / 16, Nt = D_MODEL / 16;
    int blocks = (Mt * Nt + 7) / 8;
    gemm_wmma_nt<<<dim3(blocks, 1, 1), 256, 0, stream>>>(
        yn, W_out, out, ROWS, D_MODEL, D_INNER, D_INNER, D_INNER, D_MODEL,
        0LL, 0LL, 0LL);
  }
  (void)in_sizes; (void)n_in; (void)out_size; (void)ws_size;
}